// DeformConv_7619271983446
// MI455X (gfx1250) — compile-verified
//
#include <hip/hip_runtime.h>
#include <math.h>

typedef __attribute__((ext_vector_type(2))) float v2f;
typedef __attribute__((ext_vector_type(8))) float v8f;

#define B_   4
#define CH   64
#define H_   128
#define W_   128
#define HW   16384          // H_*W_
#define KK   9
#define KD   576            // CH*KK
#define NPIX 65536          // B_*HW

// ---------------------------------------------------------------------------
// Kernel 1: offset conv (3x3, 64 -> 27) + decode dy/dx/mask
// ---------------------------------------------------------------------------
__global__ __launch_bounds__(256) void offsets_kernel(
    const float* __restrict__ x, const float* __restrict__ w_off,
    const float* __restrict__ b_off,
    float* __restrict__ dy, float* __restrict__ dx, float* __restrict__ mk)
{
    int p = blockIdx.x * 256 + threadIdx.x;   // 0 .. NPIX-1
    int b = p >> 14;
    int s = p & (HW - 1);
    int h = s >> 7;
    int w = s & (W_ - 1);

    const float* xb = x + (size_t)b * CH * HW;

    float acc[27];
#pragma unroll
    for (int i = 0; i < 27; i++) acc[i] = b_off[i];

    for (int c = 0; c < CH; c++) {
        const float* xc = xb + c * HW;
        float xv[9];
#pragma unroll
        for (int j = 0; j < 9; j++) {
            int yy = h - 1 + j / 3;
            int xx = w - 1 + j % 3;
            bool ok = (yy >= 0) & (yy < H_) & (xx >= 0) & (xx < W_);
            xv[j] = ok ? xc[yy * W_ + xx] : 0.0f;
        }
        const float* wo = w_off + c * 9;    // uniform -> scalar loads
#pragma unroll
        for (int oc = 0; oc < 27; oc++) {
#pragma unroll
            for (int j = 0; j < 9; j++)
                acc[oc] = fmaf(xv[j], wo[oc * KD + j], acc[oc]);
        }
    }
    // concat([o1,o2]) is just channels 0..17 -> dy[j]=acc[2j], dx[j]=acc[2j+1]
#pragma unroll
    for (int j = 0; j < 9; j++) {
        int g = (b * 9 + j) * HW + s;
        dy[g] = acc[2 * j];
        dx[g] = acc[2 * j + 1];
        mk[g] = 1.0f / (1.0f + __expf(-acc[18 + j]));
    }
}

// ---------------------------------------------------------------------------
// Kernel 2: transpose w (CHO x 576) -> wT (576 x CHO) for coalesced B loads
// ---------------------------------------------------------------------------
__global__ __launch_bounds__(256) void transpose_w_kernel(
    const float* __restrict__ w, float* __restrict__ wT)
{
    int i = blockIdx.x * 256 + threadIdx.x;   // 0 .. KD*CH-1
    if (i < KD * CH) {
        int k = i >> 6;       // /64
        int n = i & 63;
        wT[k * CH + n] = w[n * KD + k];
    }
}

// ---------------------------------------------------------------------------
// Kernel 3: deformable gather + fp32 WMMA GEMM (M=16 px, N=64 ch, K=576)
//   block = 128 threads = 4 waves; wave i handles channel column i*16..+15
// ---------------------------------------------------------------------------
__global__ __launch_bounds__(128) void deform_wmma_kernel(
    const float* __restrict__ x,
    const float* __restrict__ dy, const float* __restrict__ dx,
    const float* __restrict__ mk,
    const float* __restrict__ wT, const float* __restrict__ bias,
    float* __restrict__ y)
{
    __shared__ float Alds[KD * 16];     // 36 KB A-panel, k-major stride 16
    __shared__ int   idxl[144 * 4];     // 4 clamped corner indices / item
    __shared__ float wgtl[144 * 4];     // 4 mask-folded bilinear weights

    const int t  = threadIdx.x;
    const int p0 = blockIdx.x * 16;     // 16 consecutive pixels, same row
    const int b  = p0 >> 14;
    const int s0 = p0 & (HW - 1);
    const int h  = s0 >> 7;
    const int w0 = s0 & (W_ - 1);

    // ---- Phase A: 144 sample descriptors (item = kk*16 + m) ----
    for (int it = t; it < 144; it += 128) {
        int kk = it >> 4;
        int m  = it & 15;
        int wc = w0 + m;
        int g  = (b * 9 + kk) * HW + (h << 7) + wc;
        float dyy = dy[g], dxx = dx[g], mm = mk[g];
        float py = (float)(h  - 1 + kk / 3) + dyy;
        float px = (float)(wc - 1 + kk % 3) + dxx;
        float y0f = floorf(py), x0f = floorf(px);
        float wy = py - y0f,    wx = px - x0f;
        int   y0 = (int)y0f,    x0 = (int)x0f;
        float wv[4] = { (1.f - wy) * (1.f - wx) * mm, (1.f - wy) * wx * mm,
                        wy * (1.f - wx) * mm,         wy * wx * mm };
#pragma unroll
        for (int ci = 0; ci < 4; ci++) {
            int yy = y0 + (ci >> 1);
            int xx = x0 + (ci & 1);
            bool ok = (yy >= 0) & (yy < H_) & (xx >= 0) & (xx < W_);
            int yc = min(max(yy, 0), H_ - 1);
            int xc = min(max(xx, 0), W_ - 1);
            idxl[it * 4 + ci] = yc * W_ + xc;
            wgtl[it * 4 + ci] = ok ? wv[ci] : 0.0f;
        }
    }
    __syncthreads();

    // ---- Phase B: build A-panel, Alds[(c*9+kk)*16 + m] ----
    const float* xb = x + (size_t)b * CH * HW;
#pragma unroll 4
    for (int i = 0; i < 72; i++) {
        int u  = t + (i << 7);          // 0 .. 9215
        int m  = u & 15;
        int r  = u >> 4;                // c*9 + kk
        int kk = r % 9;
        int c  = r / 9;
        int it = (kk << 4) + m;
        const float* xc = xb + c * HW;
        float v = wgtl[it * 4 + 0] * xc[idxl[it * 4 + 0]]
                + wgtl[it * 4 + 1] * xc[idxl[it * 4 + 1]]
                + wgtl[it * 4 + 2] * xc[idxl[it * 4 + 2]]
                + wgtl[it * 4 + 3] * xc[idxl[it * 4 + 3]];
        Alds[u] = v;                    // coalesced LDS store
    }
    __syncthreads();

    // ---- Phase C: 144 x v_wmma_f32_16x16x4_f32 ----
    const int lane = t & 31;
    const int lo   = (lane < 16) ? 0 : 2;   // K sub-offset per ISA A layout
    const int m_   = lane & 15;
    const int n_   = ((t >> 5) << 4) + (lane & 15);   // wave's channel column

    v8f acc = {};
#pragma unroll 8
    for (int k = 0; k < KD; k += 4) {
        v2f a, bb;
        a.x  = Alds[(k + lo)     * 16 + m_];
        a.y  = Alds[(k + lo + 1) * 16 + m_];
        bb.x = wT[(k + lo)     * CH + n_];
        bb.y = wT[(k + lo + 1) * CH + n_];
        acc = __builtin_amdgcn_wmma_f32_16x16x4_f32(
                  false, a, false, bb, (short)0, acc, false, false);
    }
    __syncthreads();   // done reading Alds; reuse it as store-staging

    // ---- stage C (16x16 f32 layout: VGPR j = row j / j+8) + bias ----
    float bn = bias[n_];
#pragma unroll
    for (int j = 0; j < 8; j++) {
        int mo = j + ((lane < 16) ? 0 : 8);
        Alds[n_ * 16 + mo] = acc[j] + bn;
    }
    __syncthreads();

    // ---- coalesced NCHW store of the 64x16 tile ----
#pragma unroll
    for (int i = 0; i < 8; i++) {
        int idx = t + (i << 7);         // 0..1023, n = idx/16, m = idx%16
        int n = idx >> 4, m = idx & 15;
        y[((size_t)b * CH + n) * HW + s0 + m] = Alds[idx];
    }
}

// ---------------------------------------------------------------------------
// Kernel 4: per-channel BN stats -> scale/shift (folds gamma/beta/eps)
// ---------------------------------------------------------------------------
__global__ __launch_bounds__(256) void bn_stats_kernel(
    const float* __restrict__ y, const float* __restrict__ gamma,
    const float* __restrict__ beta, float* __restrict__ ss)
{
    int ch = blockIdx.x;
    int t  = threadIdx.x;
    float s = 0.f, s2 = 0.f;
    for (int i = t; i < B_ * HW; i += 256) {
        float v = y[((size_t)(i >> 14) * CH + ch) * HW + (i & (HW - 1))];
        s += v; s2 += v * v;
    }
    __shared__ float rs[256], rs2[256];
    rs[t] = s; rs2[t] = s2;
    __syncthreads();
    for (int o = 128; o > 0; o >>= 1) {
        if (t < o) { rs[t] += rs[t + o]; rs2[t] += rs2[t + o]; }
        __syncthreads();
    }
    if (t == 0) {
        const float inv = 1.0f / (float)(B_ * HW);
        float mean = rs[0] * inv;
        float var  = rs2[0] * inv - mean * mean;
        float sc   = gamma[ch] * rsqrtf(var + 1e-5f);
        ss[ch * 2]     = sc;
        ss[ch * 2 + 1] = beta[ch] - mean * sc;
    }
}

// ---------------------------------------------------------------------------
// Kernel 5: in-place normalize + ReLU (float4 vectorized)
// ---------------------------------------------------------------------------
__global__ __launch_bounds__(256) void bn_apply_kernel(
    float* __restrict__ y, const float* __restrict__ ss)
{
    int i = blockIdx.x * 256 + threadIdx.x;   // one float4 per thread
    int e = i << 2;
    int ch = (e >> 14) & (CH - 1);
    float sc = ss[ch * 2], sh = ss[ch * 2 + 1];
    float4 v = *reinterpret_cast<float4*>(y + e);
    v.x = fmaxf(0.f, fmaf(v.x, sc, sh));
    v.y = fmaxf(0.f, fmaf(v.y, sc, sh));
    v.z = fmaxf(0.f, fmaf(v.z, sc, sh));
    v.w = fmaxf(0.f, fmaf(v.w, sc, sh));
    *reinterpret_cast<float4*>(y + e) = v;
}

// ---------------------------------------------------------------------------
extern "C" void kernel_launch(void* const* d_in, const int* in_sizes, int n_in,
                              void* d_out, int out_size, void* d_ws, size_t ws_size,
                              hipStream_t stream)
{
    const float* x     = (const float*)d_in[0];
    const float* w_off = (const float*)d_in[1];
    const float* b_off = (const float*)d_in[2];
    const float* w     = (const float*)d_in[3];
    const float* bias  = (const float*)d_in[4];
    const float* gamma = (const float*)d_in[5];
    const float* beta  = (const float*)d_in[6];
    float* out = (float*)d_out;                 // reused as pre-BN y buffer

    // workspace layout (floats): dy | dx | mask | wT | scale-shift
    float* ws  = (float*)d_ws;
    float* dy  = ws;
    float* dx  = dy + (size_t)B_ * 9 * HW;
    float* mk  = dx + (size_t)B_ * 9 * HW;
    float* wT  = mk + (size_t)B_ * 9 * HW;
    float* ss  = wT + (size_t)KD * CH;

    offsets_kernel   <<<NPIX / 256, 256, 0, stream>>>(x, w_off, b_off, dy, dx, mk);
    transpose_w_kernel<<<(KD * CH + 255) / 256, 256, 0, stream>>>(w, wT);
    deform_wmma_kernel<<<NPIX / 16, 128, 0, stream>>>(x, dy, dx, mk, wT, bias, out);
    bn_stats_kernel  <<<CH, 256, 0, stream>>>(out, gamma, beta, ss);
    bn_apply_kernel  <<<(B_ * CH * HW) / (256 * 4), 256, 0, stream>>>(out, ss);
}